// ExpMapFlow_36352603193792
// MI455X (gfx1250) — compile-verified
//
#include <hip/hip_runtime.h>
#include <hip/hip_bf16.h>

#define STRIDE 65   // row stride for 64x64 LDS tiles (bank-conflict padding)

typedef float v2f __attribute__((ext_vector_type(2)));
typedef float v8f __attribute__((ext_vector_type(8)));

// Out[a][b] = sum_k X[k][a] * scale_k * X[k][b]   (64x64 Gram, f32 WMMA 16x16x4)
// X, Out are 64x64 with row stride STRIDE in LDS. 256 threads = 8 wave32s,
// each wave computes 2 of the 16 output 16x16 tiles, 16 K-steps each.
template <bool SCALED>
__device__ inline void gram_wmma(const float* __restrict__ X,
                                 const float* __restrict__ scale,
                                 float* __restrict__ Out, int tid) {
  const int wave = tid >> 5;
  const int lane = tid & 31;
  const int m    = lane & 15;   // M (A) == N (B) == output col within tile
  const int h    = lane >> 4;   // K-half selector
  for (int t = wave; t < 16; t += 8) {
    const int I = t >> 2, J = t & 3;
    v8f acc = {0.f, 0.f, 0.f, 0.f, 0.f, 0.f, 0.f, 0.f};
    const float* xa = X + 16 * I + m;
    const float* xb = X + 16 * J + m;
    #pragma unroll
    for (int kk = 0; kk < 16; ++kk) {
      const int ka = 4 * kk + 2 * h;
      v2f a, b;
      a.x = xa[ka * STRIDE];
      a.y = xa[(ka + 1) * STRIDE];
      float s0 = SCALED ? scale[ka] : 1.0f;
      float s1 = SCALED ? scale[ka + 1] : 1.0f;
      b.x = s0 * xb[ka * STRIDE];
      b.y = s1 * xb[(ka + 1) * STRIDE];
      acc = __builtin_amdgcn_wmma_f32_16x16x4_f32(
          /*neg_a=*/false, a, /*neg_b=*/false, b,
          /*c_mod=*/(short)0, acc, /*reuse_a=*/false, /*reuse_b=*/false);
    }
    #pragma unroll
    for (int r = 0; r < 8; ++r) {
      const int row = 16 * I + r + 8 * h;
      const int col = 16 * J + m;
      Out[row * STRIDE + col] = acc[r];
    }
  }
}

__global__ __launch_bounds__(256) void expmap_flow_kernel(
    const float* __restrict__ xs, const float* __restrict__ mu,
    const float* __restrict__ alpha, const float* __restrict__ beta,
    float* __restrict__ out_z, float* __restrict__ out_logdet,
    float* __restrict__ out_sign) {
  __shared__ float sBufA[64 * STRIDE];  // mu  -> later G = JE^T JE
  __shared__ float sBufB[64 * STRIDE];  // A1  -> J -> JE
  __shared__ float sx[64], sal[64], sbe[64];
  __shared__ float st[64], sp[64], sc[64], spb[64];
  __shared__ float sg[64], shx[64], sv[64], sq[64];
  __shared__ float su[64], sw[64], suh[64], sJuh[64];
  __shared__ float sS[16];

  const int pt  = blockIdx.x;
  const int tid = threadIdx.x;

  // ---- load mu (shared across points, L2-resident), x, alpha, beta ----
  for (int e = tid; e < 64 * 64; e += 256) {
    const int k = e >> 6, i = e & 63;
    sBufA[k * STRIDE + i] = mu[e];
  }
  if (tid < 64) {
    sx[tid]  = xs[pt * 64 + tid];
    sal[tid] = alpha[tid];
    sbe[tid] = beta[tid];
  }
  __syncthreads();

  // ---- t_k = mu_k . x ; scores ----
  if (tid < 64) {
    float acc = 0.f;
    const float* row = sBufA + tid * STRIDE;
    for (int i = 0; i < 64; ++i) acc += row[i] * sx[i];
    st[tid] = acc;
    sp[tid] = sbe[tid] * acc + sal[tid];  // score s_k
  }
  __syncthreads();
  if (tid == 0) {
    float mx = sp[0];
    for (int k = 1; k < 64; ++k) mx = fmaxf(mx, sp[k]);
    sS[0] = mx;
  }
  __syncthreads();
  if (tid < 64) sp[tid] = expf(sp[tid] - sS[0]);
  __syncthreads();
  if (tid == 0) {
    float sum = 0.f;
    for (int k = 0; k < 64; ++k) sum += sp[k];
    sS[1] = 1.0f / sum;
  }
  __syncthreads();
  if (tid < 64) {
    const float p = sp[tid] * sS[1];
    sp[tid]  = p;
    spb[tid] = p * sbe[tid];             // p*beta
    sc[tid]  = p * sbe[tid] * sbe[tid];  // p*beta^2
  }
  __syncthreads();

  // ---- g = M^T (p*beta) ; hxm = M^T (c .* t) ----
  if (tid < 64) {
    float gacc = 0.f, hacc = 0.f;
    for (int k = 0; k < 64; ++k) {
      const float mki = sBufA[k * STRIDE + tid];
      gacc += spb[k] * mki;
      hacc += sc[k] * st[k] * mki;
    }
    sg[tid]  = gacc;
    shx[tid] = hacc;
  }
  __syncthreads();
  if (tid == 0) {
    float xg = 0.f;
    for (int i = 0; i < 64; ++i) xg += sx[i] * sg[i];
    sS[2] = xg;
  }
  __syncthreads();
  const float xg = sS[2];
  if (tid < 64) {
    sv[tid] = sg[tid] - xg * sx[tid];                // v (tangent grad)
    sq[tid] = sg[tid] + (shx[tid] - xg * sg[tid]);   // q = g + Hx
  }
  __syncthreads();
  if (tid == 0) {
    float nn = 0.f, gv = 0.f, xv = 0.f;
    for (int i = 0; i < 64; ++i) {
      nn += sv[i] * sv[i];
      gv += sg[i] * sv[i];
      xv += sx[i] * sv[i];
    }
    const float n  = sqrtf(nn);
    const float ns = fmaxf(n, 1e-12f);
    const float cN = cosf(n), sN = sinf(n);
    sS[3] = ns; sS[4] = cN; sS[5] = sN;
    sS[6] = sN / ns;                        // sin(n)/n
    sS[7] = cN / ns - sN / (ns * ns);       // coef of v in u
    sS[8] = gv; sS[9] = xv;
  }
  __syncthreads();
  const float ns = sS[3], cN = sS[4], sN = sS[5];
  const float sOs = sS[6], cvf = sS[7], gv = sS[8], xv = sS[9];
  if (tid < 64) {
    su[tid] = -sN * sx[tid] + cvf * sv[tid];
    out_z[pt * 64 + tid] = cN * sx[tid] + sOs * sv[tid];  // z output
  }
  __syncthreads();

  // ---- A1 = M^T diag(c) M  (WMMA f32 Gram) ----
  gram_wmma<true>(sBufA, sc, sBufB, tid);
  __syncthreads();

  // ---- w = (A1 v - gv*g - xv*q - xg*v) / n ----
  if (tid < 64) {
    float acc = 0.f;
    const float* row = sBufB + tid * STRIDE;  // A1 symmetric: row == col
    for (int j = 0; j < 64; ++j) acc += row[j] * sv[j];
    sw[tid] = (acc - gv * sg[tid] - xv * sq[tid] - xg * sv[tid]) / ns;
  }
  __syncthreads();

  // ---- J in place: J = cN*I + sOs*(A1 - g g^T - x q^T - xg I) + u w^T ----
  for (int e = tid; e < 64 * 64; e += 256) {
    const int i = e >> 6, j = e & 63;
    float a = sBufB[i * STRIDE + j];
    a -= sg[i] * sg[j] + sx[i] * sq[j];
    if (i == j) a -= xg;
    float Jij = sOs * a + su[i] * sw[j];
    if (i == j) Jij += cN;
    sBufB[i * STRIDE + j] = Jij;
  }
  __syncthreads();

  // ---- Householder direction u_h = normalize(x + sgn * e_63) ----
  if (tid == 0) {
    const float sgn = (sx[63] >= 0.f) ? 1.f : -1.f;
    float nrm = 0.f;
    for (int i = 0; i < 64; ++i) {
      const float uvi = sx[i] + ((i == 63) ? sgn : 0.f);
      suh[i] = uvi;
      nrm += uvi * uvi;
    }
    sS[10] = 1.0f / sqrtf(nrm);
  }
  __syncthreads();
  if (tid < 64) suh[tid] *= sS[10];
  __syncthreads();

  // ---- Juh = J @ u_h ----
  if (tid < 64) {
    float acc = 0.f;
    const float* row = sBufB + tid * STRIDE;
    for (int j = 0; j < 64; ++j) acc += row[j] * suh[j];
    sJuh[tid] = acc;
  }
  __syncthreads();

  // ---- JE in place (E from Householder; rank-1 update; pad col 63 = 0) ----
  for (int e = tid; e < 64 * 64; e += 256) {
    const int i = e >> 6, j = e & 63;
    float val = sBufB[i * STRIDE + j] - 2.f * sJuh[i] * suh[j];
    if (j == 63) val = 0.f;
    sBufB[i * STRIDE + j] = val;
  }
  __syncthreads();

  // ---- G = JE^T JE (WMMA f32 Gram), overwrites mu buffer ----
  gram_wmma<false>(sBufB, nullptr, sBufA, tid);
  __syncthreads();

  // ---- Cholesky of SPD 63x63 block; 0.5*logdet = sum log L_kk; sign = +1 ----
  float ld = 0.f;
  for (int k = 0; k < 63; ++k) {
    if (tid == 0) {
      const float d = sBufA[k * STRIDE + k];
      const float L = sqrtf(d);
      sBufA[k * STRIDE + k] = L;
      ld += logf(L);
      sS[12] = 1.0f / L;
    }
    __syncthreads();
    const int mrem = 62 - k;  // rows strictly below diagonal
    if (tid < mrem) {
      const int i = k + 1 + tid;
      sBufA[i * STRIDE + k] *= sS[12];
    }
    __syncthreads();
    for (int e = tid; e < mrem * mrem; e += 256) {
      const int i = k + 1 + e / mrem;
      const int j = k + 1 + e % mrem;
      if (j <= i)
        sBufA[i * STRIDE + j] -= sBufA[i * STRIDE + k] * sBufA[j * STRIDE + k];
    }
    __syncthreads();
  }
  if (tid == 0) {
    out_logdet[pt] = ld;
    out_sign[pt]   = 1.0f;  // Gram matrix is SPD
  }
}

extern "C" void kernel_launch(void* const* d_in, const int* in_sizes, int n_in,
                              void* d_out, int out_size, void* d_ws, size_t ws_size,
                              hipStream_t stream) {
  const float* xs    = (const float*)d_in[0];
  const float* mu    = (const float*)d_in[1];
  const float* alpha = (const float*)d_in[2];
  const float* beta  = (const float*)d_in[3];
  const int N = in_sizes[0] / 64;  // 1024
  float* out       = (float*)d_out;
  float* out_z     = out;                // N*64
  float* out_logd  = out + N * 64;       // N
  float* out_sign  = out + N * 64 + N;   // N
  expmap_flow_kernel<<<dim3(N), dim3(256), 0, stream>>>(
      xs, mu, alpha, beta, out_z, out_logd, out_sign);
}